// Aleat5_1ChamferL2NormalWeightedALLSVDLoss_87222195847750
// MI455X (gfx1250) — compile-verified
//
#include <hip/hip_runtime.h>
#include <math.h>

#define NPTS 20000
#define NB   4
#define BIGF 1e30f
#define CHSZ 2000   // target chunk staged in LDS (125 groups of 16; 25 groups of 80)

typedef __attribute__((ext_vector_type(2))) float v2f;
typedef __attribute__((ext_vector_type(8))) float v8f;

// ---------------------------------------------------------------------------
// Kernel 1: per-batch bounding box -> block bounds (BLOCK_IDX = 0)
// box layout per batch (stride 8 floats): [bmin.xyz, pad, bmax.xyz, pad]
// ---------------------------------------------------------------------------
__global__ void __launch_bounds__(256) box_kernel(const float* __restrict__ pred,
                                                  float* __restrict__ box) {
    const int b = blockIdx.x;
    const int tid = threadIdx.x;
    __shared__ float smn[3][256];
    __shared__ float smx[3][256];

    float mn[3] = { BIGF,  BIGF,  BIGF};
    float mx[3] = {-BIGF, -BIGF, -BIGF};
    for (int n = tid; n < NPTS; n += 256) {
        const float* p = pred + ((size_t)b * NPTS + n) * 3;
#pragma unroll
        for (int c = 0; c < 3; ++c) {
            float v = p[c];
            mn[c] = fminf(mn[c], v);
            mx[c] = fmaxf(mx[c], v);
        }
    }
#pragma unroll
    for (int c = 0; c < 3; ++c) { smn[c][tid] = mn[c]; smx[c][tid] = mx[c]; }
    __syncthreads();
    for (int s = 128; s > 0; s >>= 1) {
        if (tid < s) {
#pragma unroll
            for (int c = 0; c < 3; ++c) {
                smn[c][tid] = fminf(smn[c][tid], smn[c][tid + s]);
                smx[c][tid] = fmaxf(smx[c][tid], smx[c][tid + s]);
            }
        }
        __syncthreads();
    }
    if (tid == 0) {
#pragma unroll
        for (int c = 0; c < 3; ++c) {
            float lo0 = smn[c][0], hi0 = smx[c][0];
            float w   = hi0 - lo0;
            float lo  = lo0 + w * 0.05f;   // MARGIN
            float hi  = hi0 - w * 0.05f;
            float bs  = (c == 0) ? 0.1f : 1.0f;  // BLOCK_SIZE
            float bmin = lo;                     // BLOCK_IDX = 0
            float bmax = bmin + (hi - lo) * bs;
            box[b * 8 + c]     = bmin;
            box[b * 8 + 4 + c] = bmax;
        }
    }
}

// ---------------------------------------------------------------------------
// Kernel 2: WMMA pairwise min-distance core.
//   d[b,m] = max( min_n ( p2[m] + t2'[n] - 2 * p.t ), 0 )
//   key    = ind_p ? d : BIG ;  indp = ind_p
// Each 256-thread block: 8 waves x 4 A-tiles x 16 rows = 512 pred rows.
// Target loop is unrolled 5x (80 targets): candidate mins form a tree of
// provably-canonical values (fma/minnum results), so clang emits only ONE
// canonicalize per accumulator per 80 targets instead of one per 16.
// ---------------------------------------------------------------------------
__global__ void __launch_bounds__(256) chamfer_kernel(const float* __restrict__ pred,
                                                      const float* __restrict__ target,
                                                      const float* __restrict__ box,
                                                      float* __restrict__ key,
                                                      unsigned* __restrict__ indp) {
    const int b    = blockIdx.y;
    const int tid  = threadIdx.x;
    const int lane = tid & 31;
    const int wv   = tid >> 5;
    const int half = lane >> 4;   // 0: K0/K1, 1: K2/K3(=0)
    const int l16  = lane & 15;

    __shared__ float4 tgt[CHSZ];

    const float bx0 = box[b * 8 + 0], by0 = box[b * 8 + 1], bz0 = box[b * 8 + 2];
    const float bx1 = box[b * 8 + 4], by1 = box[b * 8 + 5], bz1 = box[b * 8 + 6];

    const int wave_m0 = blockIdx.x * 512 + wv * 64;

    // A operands: ISA 32-bit 16x4 layout. Lane l holds row M = l%16;
    // lanes 0-15 carry (K0,K1)=(x,y), lanes 16-31 carry (K2,K3)=(z,0).
    v2f a[4];
#pragma unroll
    for (int t = 0; t < 4; ++t) {
        int m  = wave_m0 + t * 16 + l16;
        int mc = m < NPTS ? m : NPTS - 1;       // clamp tail (EXEC stays full)
        const float* p = pred + ((size_t)b * NPTS + mc) * 3;
        float px = p[0], py = p[1], pz = p[2];
        v2f av;
        av.x = half ? pz : px;
        av.y = half ? 0.0f : py;
        a[t] = av;
    }

    // Running min of (t2' - 2*dot) per C/D element.
    float cm[4][8];
#pragma unroll
    for (int t = 0; t < 4; ++t)
#pragma unroll
        for (int r = 0; r < 8; ++r) cm[t][r] = BIGF;

    const int nchunks = NPTS / CHSZ;
    for (int ch = 0; ch < nchunks; ++ch) {
        const int base = ch * CHSZ;
        for (int i = tid; i < CHSZ; i += 256) {
            const float* tp = target + ((size_t)b * NPTS + base + i) * 3;
            float tx = tp[0], ty = tp[1], tz = tp[2];
            float t2 = tx * tx + ty * ty + tz * tz;
            bool valid = (bx0 < tx) && (tx < bx1) && (by0 < ty) && (ty < by1) &&
                         (bz0 < tz) && (tz < bz1);
            tgt[i] = make_float4(tx, ty, tz, valid ? t2 : t2 + BIGF);
        }
        // speculative prefetch of next chunk (global_prefetch_b8)
        if (ch + 1 < nchunks) {
            int pn = (ch + 1) * CHSZ + tid * 8;
            if (pn >= NPTS) pn = NPTS - 1;
            __builtin_prefetch(target + ((size_t)b * NPTS + pn) * 3, 0, 1);
        }
        __syncthreads();

        for (int j = 0; j < CHSZ; j += 80) {
            // 5 B operands (16 targets each) for this unrolled group.
            v2f   bm[5];
            float t2l[5];
#pragma unroll
            for (int s = 0; s < 5; ++s) {
                float4 tv = tgt[j + s * 16 + l16];
                bm[s].x = half ? tv.z : tv.x;
                bm[s].y = half ? 0.0f : tv.y;
                t2l[s]  = tv.w;   // per-lane: column N = l16
            }
#pragma unroll
            for (int t = 0; t < 4; ++t) {
                float tmp[8];
#pragma unroll
                for (int s = 0; s < 5; ++s) {
                    v8f czero = {};
                    v8f dc = __builtin_amdgcn_wmma_f32_16x16x4_f32(
                        /*neg_a=*/false, a[t], /*neg_b=*/false, bm[s],
                        /*c_mod=*/(short)0, czero,
                        /*reuse_a=*/false, /*reuse_b=*/false);
#pragma unroll
                    for (int r = 0; r < 8; ++r) {
                        float cand = fmaf(-2.0f, dc[r], t2l[s]);
                        tmp[r] = (s == 0) ? cand : fminf(tmp[r], cand);
                    }
                }
                // Single PHI-merge (one canonicalize) per element per 80 targets.
#pragma unroll
                for (int r = 0; r < 8; ++r)
                    cm[t][r] = fminf(cm[t][r], tmp[r]);
            }
        }
        __syncthreads();
    }

    // Min across the 16 column-lanes of each half (butterfly stays within half).
#pragma unroll
    for (int t = 0; t < 4; ++t)
#pragma unroll
        for (int r = 0; r < 8; ++r) {
            float v = cm[t][r];
            v = fminf(v, __shfl_xor(v, 1, 32));
            v = fminf(v, __shfl_xor(v, 2, 32));
            v = fminf(v, __shfl_xor(v, 4, 32));
            v = fminf(v, __shfl_xor(v, 8, 32));
            cm[t][r] = v;
        }

    // Lanes 0 (rows r) and 16 (rows r+8) write results.
    if (l16 == 0) {
#pragma unroll
        for (int t = 0; t < 4; ++t)
#pragma unroll
            for (int r = 0; r < 8; ++r) {
                int M = wave_m0 + t * 16 + r + half * 8;
                if (M < NPTS) {
                    const float* p = pred + ((size_t)b * NPTS + M) * 3;
                    float px = p[0], py = p[1], pz = p[2];
                    float p2 = px * px + py * py + pz * pz;
                    float d  = fmaxf(p2 + cm[t][r], 0.0f);
                    bool sel = (bx0 < px) && (px < bx1) && (by0 < py) &&
                               (py < by1) && (bz0 < pz) && (pz < bz1);
                    key[(size_t)b * NPTS + M]  = sel ? d : BIGF;
                    indp[(size_t)b * NPTS + M] = sel ? 1u : 0u;
                }
            }
    }
}

// ---------------------------------------------------------------------------
// Kernel 3: per-batch exact kth value (radix select on float bits, keys >= 0)
// then masked sums -> loss_b[b].
// ---------------------------------------------------------------------------
__global__ void __launch_bounds__(256) select_kernel(const float* __restrict__ key,
                                                     const unsigned* __restrict__ indp,
                                                     const float* __restrict__ mask,
                                                     float* __restrict__ lossb) {
    const int b   = blockIdx.x;
    const int tid = threadIdx.x;
    const float*    kb = key  + (size_t)b * NPTS;
    const unsigned* ib = indp + (size_t)b * NPTS;
    const float*    mb = mask + (size_t)b * NPTS;

    __shared__ unsigned hist[16];
    __shared__ unsigned ured[256];
    __shared__ float    fred[256];
    __shared__ float    fred2[256];
    __shared__ unsigned uprefix;
    __shared__ int      kk;

    // count = sum(ind_p); kidx = int(float(count) * RATIO)
    unsigned cnt = 0;
    for (int n = tid; n < NPTS; n += 256) cnt += ib[n];
    ured[tid] = cnt;
    __syncthreads();
    for (int s = 128; s > 0; s >>= 1) {
        if (tid < s) ured[tid] += ured[tid + s];
        __syncthreads();
    }
    if (tid == 0) {
        int count = (int)ured[0];
        kk = (int)((float)count * 0.5f);
        uprefix = 0u;
    }
    __syncthreads();

    // 8-pass MSB-first 4-bit radix select over bits of where(ind_p, d, BIG)
    for (int shift = 28; shift >= 0; shift -= 4) {
        if (tid < 16) hist[tid] = 0u;
        __syncthreads();
        unsigned pmask = (shift == 28) ? 0u : (0xFFFFFFFFu << (shift + 4));
        unsigned pref  = uprefix & pmask;
        for (int n = tid; n < NPTS; n += 256) {
            unsigned u = __float_as_uint(kb[n]);
            if ((u & pmask) == pref) atomicAdd(&hist[(u >> shift) & 15], 1u);
        }
        __syncthreads();
        if (tid == 0) {
            unsigned cum = 0;
            int k = kk;
            for (int dgt = 0; dgt < 16; ++dgt) {
                unsigned h = hist[dgt];
                if ((unsigned)k < cum + h) {
                    uprefix |= ((unsigned)dgt) << shift;
                    kk = k - (int)cum;
                    break;
                }
                cum += h;
            }
        }
        __syncthreads();
    }
    const float m = __uint_as_float(uprefix);

    // keep = ind_p & (d < m);  loss_b = sum(d*d*mk) / (sum(mk) + 1e-12)
    float ssq = 0.0f, scn = 0.0f;
    for (int n = tid; n < NPTS; n += 256) {
        float d = kb[n];
        if (ib[n] && d < m) {
            float mk = mb[n];
            ssq += d * d * mk;
            scn += mk;
        }
    }
    fred[tid] = ssq;
    fred2[tid] = scn;
    __syncthreads();
    for (int s = 128; s > 0; s >>= 1) {
        if (tid < s) { fred[tid] += fred[tid + s]; fred2[tid] += fred2[tid + s]; }
        __syncthreads();
    }
    if (tid == 0) lossb[b] = fred[0] / (fred2[0] + 1e-12f);
}

// ---------------------------------------------------------------------------
// Kernel 4: mean over batch + GAMMA=0 focal/aleatoric epilogue.
// ---------------------------------------------------------------------------
__global__ void final_kernel(const float* __restrict__ lossb,
                             const float* __restrict__ alpha,
                             float* __restrict__ out) {
    if (threadIdx.x == 0 && blockIdx.x == 0) {
        float loss = 0.25f * (lossb[0] + lossb[1] + lossb[2] + lossb[3]);
        float a = alpha[0];
        float e = expf(-a);
        float x = e * loss;
        float fw = powf(x, 0.0f);        // GAMMA = 0
        fw = fw / (fw + 1e-12f);         // fw / (sum(fw) + 1e-12), fw shape [1]
        out[0] = fw * x + a;
    }
}

// ---------------------------------------------------------------------------
extern "C" void kernel_launch(void* const* d_in, const int* in_sizes, int n_in,
                              void* d_out, int out_size, void* d_ws, size_t ws_size,
                              hipStream_t stream) {
    const float* pred   = (const float*)d_in[0];
    const float* target = (const float*)d_in[1];
    const float* mask   = (const float*)d_in[2];
    const float* alpha  = (const float*)d_in[3];
    float* out = (float*)d_out;

    // workspace layout (floats): [0..31] box, [32..35] loss_b, [64..] key, then indp
    float* box   = (float*)d_ws;
    float* lossb = box + 32;
    float* key   = box + 64;
    unsigned* indp = (unsigned*)(key + (size_t)NB * NPTS);

    box_kernel<<<NB, 256, 0, stream>>>(pred, box);
    chamfer_kernel<<<dim3((NPTS + 511) / 512, NB), 256, 0, stream>>>(pred, target, box,
                                                                     key, indp);
    select_kernel<<<NB, 256, 0, stream>>>(key, indp, mask, lossb);
    final_kernel<<<1, 32, 0, stream>>>(lossb, alpha, out);
    (void)in_sizes; (void)n_in; (void)out_size; (void)ws_size;
}